// InterDownGraph_17068200034582
// MI455X (gfx1250) — compile-verified
//
#include <hip/hip_runtime.h>
#include <hip/hip_bf16.h>

// ---- problem constants (match reference) ----
#define NPT      16384   // N points per batch
#define NPOINT   1024    // FPS samples
#define KNN      20      // neighbors kept (reference drops self from top-21)
#define TILE     16      // WMMA tile edge

typedef __attribute__((ext_vector_type(2))) float v2f;
typedef __attribute__((ext_vector_type(8))) float v8f;

// =====================================================================
// Kernel 1: furthest point sampling. One 1024-thread block per batch.
// Each thread owns 16 points cached in VGPRs. Per serial step:
//   - per-thread argmax over its 16 running min-dists
//   - wave32 shuffle argmax (5 steps, no barrier)
//   - 32 wave winners -> LDS -> wave 0 shuffle argmax
// Total order (dist desc, index asc) reproduces jnp.argmax tie-break.
// =====================================================================
#define FPS_THREADS 1024
#define PPT (NPT / FPS_THREADS)   // 16 points per thread

__device__ __forceinline__ void argmax_combine(float& d, int& i, float od, int oi) {
    if (od > d || (od == d && oi < i)) { d = od; i = oi; }
}

__global__ void fps_kernel(const float* __restrict__ pts,
                           int* __restrict__ out_ind,      // [B, NPOINT]
                           float* __restrict__ out_query)  // [B, NPOINT, 3]
{
    const int b = blockIdx.x;
    const int t = threadIdx.x;
    const int wave = t >> 5;
    const int lane = t & 31;
    const float* p = pts + (size_t)b * NPT * 3;

    float px[PPT], py[PPT], pz[PPT], md[PPT];
#pragma unroll
    for (int k = 0; k < PPT; ++k) {
        int i = t + k * FPS_THREADS;
        px[k] = p[i * 3 + 0];
        py[k] = p[i * 3 + 1];
        pz[k] = p[i * 3 + 2];
        md[k] = 1e10f;
    }

    __shared__ float wrd[FPS_THREADS / 32];
    __shared__ int   wri[FPS_THREADS / 32];
    __shared__ float lastp[3];

    if (t == 0) {
        out_ind[(size_t)b * NPOINT] = 0;
        out_query[((size_t)b * NPOINT) * 3 + 0] = p[0];
        out_query[((size_t)b * NPOINT) * 3 + 1] = p[1];
        out_query[((size_t)b * NPOINT) * 3 + 2] = p[2];
        lastp[0] = p[0]; lastp[1] = p[1]; lastp[2] = p[2];
    }
    __syncthreads();

    for (int s = 1; s < NPOINT; ++s) {
        const float lx = lastp[0], ly = lastp[1], lz = lastp[2];
        float bestd = -1.0f;
        int   besti = 0;
#pragma unroll
        for (int k = 0; k < PPT; ++k) {
            float dx = px[k] - lx, dy = py[k] - ly, dz = pz[k] - lz;
            float d  = dx * dx + dy * dy + dz * dz;
            float m  = fminf(md[k], d);
            md[k] = m;
            if (m > bestd) { bestd = m; besti = t + k * FPS_THREADS; }
        }

        // wave32 argmax, no barriers
#pragma unroll
        for (int o = 16; o > 0; o >>= 1) {
            float od = __shfl_down(bestd, o, 32);
            int   oi = __shfl_down(besti, o, 32);
            argmax_combine(bestd, besti, od, oi);
        }
        if (lane == 0) { wrd[wave] = bestd; wri[wave] = besti; }
        __syncthreads();

        if (wave == 0) {
            float d2 = wrd[lane];
            int   i2 = wri[lane];
#pragma unroll
            for (int o = 16; o > 0; o >>= 1) {
                float od = __shfl_down(d2, o, 32);
                int   oi = __shfl_down(i2, o, 32);
                argmax_combine(d2, i2, od, oi);
            }
            if (lane == 0) {
                out_ind[(size_t)b * NPOINT + s] = i2;
                float x = p[i2 * 3 + 0], y = p[i2 * 3 + 1], z = p[i2 * 3 + 2];
                lastp[0] = x; lastp[1] = y; lastp[2] = z;
                size_t o = ((size_t)b * NPOINT + s) * 3;
                out_query[o + 0] = x; out_query[o + 1] = y; out_query[o + 2] = z;
            }
        }
        __syncthreads();
    }
}

// =====================================================================
// Kernel 2/3: brute-force KNN (top-21, drop self) using WMMA f32 16x16x4
// for the Q.P^T Gram tile. One wave (32 lanes) per 16-query tile.
//   A (16x4): lanes 0-15 hold row M=lane, K={0,1}=(x,y); lanes 16-31 K={2,3}=(z,0)
//   B (4x16): lanes 0-15 hold col N=lane, K={0,1};       lanes 16-31 K={2,3}
//   C (16x16): lane = column N (point), VGPR r = row M = r + 8*(lane>=16)
// dist = |q|^2 + |p|^2 - 2*(q.p), clamped >= 0  (matches _pdist2).
// Selection: register-resident sorted list, branch-free insertion network
// (strict < comparisons => stable ascending-index ties, matching top_k).
// =====================================================================
__global__ void knn_kernel(const float* __restrict__ qry,   // [B, NQ, 3]
                           const float* __restrict__ pts,   // [B, NP, 3]
                           int NQ, int NP,
                           int* __restrict__ out_nbr,       // [B, NQ, KNN]
                           float* __restrict__ out_d)       // [B, NQ, KNN, 3]
{
    const int b    = blockIdx.y;
    const int tile = blockIdx.x;        // 16-query tile index
    const int lane = threadIdx.x;       // 0..31
    const int half = lane >> 4;
    const int m    = lane & 15;

    const float* q = qry + ((size_t)b * NQ + (size_t)tile * TILE) * 3;
    const float* p = pts + (size_t)b * NP * 3;

    // A operand: fixed for the whole scan
    const float qx = q[m * 3 + 0], qy = q[m * 3 + 1], qz = q[m * 3 + 2];
    v2f a;
    a.x = half ? qz : qx;
    a.y = half ? 0.0f : qy;

    __shared__ float sqn[TILE];
    __shared__ float sdist[TILE][TILE];
    if (lane < TILE) sqn[lane] = qx * qx + qy * qy + qz * qz;
    __syncthreads();

    float qn[8];
#pragma unroll
    for (int r = 0; r < 8; ++r) qn[r] = sqn[r + 8 * half];

    // per-lane sorted top-(K+1) list -- stays in VGPRs (const indices only)
    float bd[KNN + 1];
    int   bi[KNN + 1];
#pragma unroll
    for (int k = 0; k <= KNN; ++k) { bd[k] = 3.0e38f; bi[k] = 0; }

    const int nchunks = NP / TILE;
    for (int c = 0; c < nchunks; ++c) {
        const int pi = c * TILE + m;
        const float x = p[pi * 3 + 0], y = p[pi * 3 + 1], z = p[pi * 3 + 2];
        const float pn = x * x + y * y + z * z;
        v2f bm;
        bm.x = half ? z : x;
        bm.y = half ? 0.0f : y;

        v8f cacc = {};
        cacc = __builtin_amdgcn_wmma_f32_16x16x4_f32(
            /*neg_a=*/false, a, /*neg_b=*/false, bm,
            /*c_mod=*/(short)0, cacc, /*reuse_a=*/false, /*reuse_b=*/false);

#pragma unroll
        for (int r = 0; r < 8; ++r) {
            float d = qn[r] + pn - 2.0f * cacc[r];
            sdist[r + 8 * half][m] = fmaxf(d, 0.0f);
        }
        __syncthreads();

        if (lane < TILE) {
#pragma unroll 1
            for (int n = 0; n < TILE; ++n) {
                float d = sdist[lane][n];
                if (d < bd[KNN]) {               // cheap reject; rest is branch-free
                    int idx = c * TILE + n;
                    // insertion network: new bd[j] =
                    //   bd[j-1] if d < bd[j-1]; else d if d < bd[j]; else bd[j]
                    // evaluated top-down so all reads see pre-insert values.
#pragma unroll
                    for (int j = KNN; j >= 1; --j) {
                        bool up   = d < bd[j - 1];
                        bool here = d < bd[j];
                        float nd = up ? bd[j - 1] : (here ? d   : bd[j]);
                        int   ni = up ? bi[j - 1] : (here ? idx : bi[j]);
                        bd[j] = nd; bi[j] = ni;
                    }
                    bool h0 = d < bd[0];
                    bi[0] = h0 ? idx : bi[0];
                    bd[0] = h0 ? d   : bd[0];
                }
            }
        }
        __syncthreads();   // reconverge before next WMMA (EXEC must be all-1s)
    }

    if (lane < TILE) {
        const size_t qg = (size_t)b * NQ + (size_t)tile * TILE + lane;  // lane<16 => qx..qz are this query
#pragma unroll 1
        for (int k = 0; k < KNN; ++k) {
            int idx = bi[k + 1];                 // drop self (entry 0)
            out_nbr[qg * KNN + k] = idx;
            float sx = p[idx * 3 + 0], sy = p[idx * 3 + 1], sz = p[idx * 3 + 2];
            size_t o = (qg * KNN + k) * 3;
            out_d[o + 0] = qx - sx;
            out_d[o + 1] = qy - sy;
            out_d[o + 2] = qz - sz;
        }
    }
}

// =====================================================================
// Launch: out buffer layout (float32 elements, ints stored bit-exact):
//   [0]                 xyz_ind    B*1024        (int32 bits)
//   [B*1024]            xyz_query  B*1024*3
//   [B*4096]            nbr_mid    B*1024*20     (int32 bits)
//   [B*24576]           d_mid      B*1024*20*3
//   [B*86016]           nbr_out    B*1024*20     (int32 bits)
//   [B*106496]          d_out      B*1024*20*3
// =====================================================================
extern "C" void kernel_launch(void* const* d_in, const int* in_sizes, int n_in,
                              void* d_out, int out_size, void* d_ws, size_t ws_size,
                              hipStream_t stream) {
    (void)n_in; (void)out_size; (void)d_ws; (void)ws_size;

    const float* points = (const float*)d_in[0];
    const int B = in_sizes[0] / (NPT * 3);

    float* out = (float*)d_out;
    int*   xyz_ind   = (int*)out;
    float* xyz_query = out + (size_t)B * NPOINT;
    int*   nbr_mid   = (int*)(out + (size_t)B * NPOINT * 4);
    float* d_mid     = out + (size_t)B * NPOINT * 4 + (size_t)B * NPOINT * KNN;
    int*   nbr_out   = (int*)(out + (size_t)B * NPOINT * (4 + KNN + 3 * KNN));
    float* d_outv    = out + (size_t)B * NPOINT * (4 + 2 * KNN + 3 * KNN);

    // 1) FPS: one block per batch (serial chain; points stay in VGPRs)
    fps_kernel<<<dim3(B), dim3(FPS_THREADS), 0, stream>>>(points, xyz_ind, xyz_query);

    // 2) Gmid: 21-NN of the 1024 queries among all 16384 points
    knn_kernel<<<dim3(NPOINT / TILE, B), dim3(32), 0, stream>>>(
        xyz_query, points, NPOINT, NPT, nbr_mid, d_mid);

    // 3) Gout: 21-NN among the 1024 queries themselves
    knn_kernel<<<dim3(NPOINT / TILE, B), dim3(32), 0, stream>>>(
        xyz_query, xyz_query, NPOINT, NPOINT, nbr_out, d_outv);
}